// SpatialGNN_67233418051658
// MI455X (gfx1250) — compile-verified
//
#include <hip/hip_runtime.h>

#define NNODES 50000
#define NEDGES 800000

typedef __attribute__((ext_vector_type(2))) float v2f;
typedef __attribute__((ext_vector_type(8))) float v8f;

__device__ __forceinline__ v8f wmma_f32_k4(v2f a, v2f b, v8f c) {
  // D = A(16x4,f32) * B(4x16,f32) + C(16x16,f32)
  return __builtin_amdgcn_wmma_f32_16x16x4_f32(
      /*neg_a=*/false, a, /*neg_b=*/false, b,
      /*c_mod=*/(short)0, c, /*reuse_a=*/false, /*reuse_b=*/false);
}

// ---------------------------------------------------------------------------
// out[M,N] = A[M,K] @ W[K,N] + bias  (optional ReLU). One wave per 16x16 tile.
// M multiple of 16, N multiple of 16, K multiple of 4.
// ---------------------------------------------------------------------------
template <bool RELU>
__global__ __launch_bounds__(256) void gemm_bias_wmma(
    const float* __restrict__ A, const float* __restrict__ W,
    const float* __restrict__ bias, float* __restrict__ out,
    int M, int K, int N) {
  const int wave = blockIdx.x * (blockDim.x >> 5) + (threadIdx.x >> 5);
  const int ntiles = N >> 4;
  const int tm = wave / ntiles;
  const int tn = wave - tm * ntiles;
  if (tm >= (M >> 4)) return;                 // wave-uniform exit (EXEC all-1s kept)
  const int lane = threadIdx.x & 31;
  const int half = lane >> 4, lo = lane & 15;
  const int m0 = tm << 4, n0 = tn << 4;

  const float* arow = A + (size_t)(m0 + lo) * K;
  v8f acc = {};
  for (int k0 = 0; k0 < K; k0 += 4) {
    const int ka = k0 + half * 2;
    v2f a; a.x = arow[ka];                a.y = arow[ka + 1];
    v2f b; b.x = W[(size_t)ka * N + n0 + lo];
           b.y = W[(size_t)(ka + 1) * N + n0 + lo];
    acc = wmma_f32_k4(a, b, acc);
  }

  const float bv = bias[n0 + lo];
  float* orow = out + (size_t)(m0 + half * 8) * N + n0 + lo;
#pragma unroll
  for (int v = 0; v < 8; ++v) {
    float r = acc[v] + bv;
    if (RELU) r = fmaxf(r, 0.0f);
    orow[(size_t)v * N] = r;
  }
}

// ---------------------------------------------------------------------------
// out[M,N] = (agg/max(cnt,1)) @ Wl + bl + h @ Wr   (pre-layernorm SAGE combine)
// ---------------------------------------------------------------------------
__global__ __launch_bounds__(256) void sage_combine_wmma(
    const float* __restrict__ agg, const float* __restrict__ cnt,
    const float* __restrict__ h, const float* __restrict__ Wl,
    const float* __restrict__ bl, const float* __restrict__ Wr,
    float* __restrict__ out, int M, int K, int N) {
  const int wave = blockIdx.x * (blockDim.x >> 5) + (threadIdx.x >> 5);
  const int ntiles = N >> 4;
  const int tm = wave / ntiles;
  const int tn = wave - tm * ntiles;
  if (tm >= (M >> 4)) return;
  const int lane = threadIdx.x & 31;
  const int half = lane >> 4, lo = lane & 15;
  const int m0 = tm << 4, n0 = tn << 4;

  const float inv = 1.0f / fmaxf(cnt[m0 + lo], 1.0f);  // per-row mean scale
  const float* grow = agg + (size_t)(m0 + lo) * K;
  const float* hrow = h + (size_t)(m0 + lo) * K;

  v8f acc = {};
  for (int k0 = 0; k0 < K; k0 += 4) {
    const int ka = k0 + half * 2;
    v2f aa; aa.x = grow[ka] * inv;        aa.y = grow[ka + 1] * inv;
    v2f bl_f; bl_f.x = Wl[(size_t)ka * N + n0 + lo];
              bl_f.y = Wl[(size_t)(ka + 1) * N + n0 + lo];
    acc = wmma_f32_k4(aa, bl_f, acc);
    v2f ah; ah.x = hrow[ka];              ah.y = hrow[ka + 1];
    v2f br_f; br_f.x = Wr[(size_t)ka * N + n0 + lo];
              br_f.y = Wr[(size_t)(ka + 1) * N + n0 + lo];
    acc = wmma_f32_k4(ah, br_f, acc);
  }

  const float bv = bl[n0 + lo];
  float* orow = out + (size_t)(m0 + half * 8) * N + n0 + lo;
#pragma unroll
  for (int v = 0; v < 8; ++v) orow[(size_t)v * N] = acc[v] + bv;
}

// ---------------------------------------------------------------------------
// Per-node LayerNorm (+ReLU). One wave32 per node; shuffle reduction.
// ---------------------------------------------------------------------------
__global__ __launch_bounds__(256) void ln_relu(
    const float* __restrict__ in, const float* __restrict__ gamma,
    const float* __restrict__ beta, float* __restrict__ out, int D) {
  const int node = blockIdx.x * (blockDim.x >> 5) + (threadIdx.x >> 5);
  if (node >= NNODES) return;
  const int lane = threadIdx.x & 31;
  const float* row = in + (size_t)node * D;
  float s = 0.0f, ss = 0.0f;
  for (int f = lane; f < D; f += 32) { float v = row[f]; s += v; ss += v * v; }
#pragma unroll
  for (int off = 16; off; off >>= 1) {
    s += __shfl_xor(s, off, 32);
    ss += __shfl_xor(ss, off, 32);
  }
  const float mu = s / (float)D;
  const float var = ss / (float)D - mu * mu;          // ddof=0 like jnp.var
  const float rs = rsqrtf(var + 1e-5f);
  float* orow = out + (size_t)node * D;
  for (int f = lane; f < D; f += 32)
    orow[f] = fmaxf((row[f] - mu) * rs * gamma[f] + beta[f], 0.0f);
}

// ---------------------------------------------------------------------------
// Edge scatter: agg[tgt] += h[src], cnt[tgt] += 1. Thread per (edge, feature).
// ---------------------------------------------------------------------------
__global__ __launch_bounds__(256) void scatter_add(
    const int* __restrict__ src, const int* __restrict__ tgt,
    const float* __restrict__ h, float* __restrict__ agg,
    float* __restrict__ cnt, int dshift) {
  const long long idx = (long long)blockIdx.x * blockDim.x + threadIdx.x;
  const int D = 1 << dshift;
  if (idx >= ((long long)NEDGES << dshift)) return;
  const int e = (int)(idx >> dshift);
  const int f = (int)(idx & (D - 1));
  const int s = src[e], t = tgt[e];
  unsafeAtomicAdd(&agg[(size_t)t * D + f], h[(size_t)s * D + f]);
  if (f == 0) unsafeAtomicAdd(&cnt[t], 1.0f);
}

__global__ __launch_bounds__(256) void zero_f32(float* __restrict__ p, long long n) {
  const long long i = (long long)blockIdx.x * blockDim.x + threadIdx.x;
  if (i < n) p[i] = 0.0f;
}

// ---------------------------------------------------------------------------
// Final [16 -> 1] projection. One thread per node.
// ---------------------------------------------------------------------------
__global__ __launch_bounds__(256) void final_proj(
    const float* __restrict__ h3, const float* __restrict__ Wo2,
    const float* __restrict__ bo2, float* __restrict__ out) {
  const int m = blockIdx.x * blockDim.x + threadIdx.x;
  if (m >= NNODES) return;
  const float* r = h3 + (size_t)m * 16;
  float acc = bo2[0];
#pragma unroll
  for (int k = 0; k < 16; ++k) acc += r[k] * Wo2[k];
  out[m] = acc;
}

extern "C" void kernel_launch(void* const* d_in, const int* in_sizes, int n_in,
                              void* d_out, int out_size, void* d_ws, size_t ws_size,
                              hipStream_t stream) {
  const float* x    = (const float*)d_in[0];   // [50000,84]
  const int*   ei   = (const int*)d_in[1];     // [2,800000]
  const float* W_in = (const float*)d_in[2];   // [84,128]
  const float* b_in = (const float*)d_in[3];
  const float* Wl1  = (const float*)d_in[4];   // [128,64]
  const float* bl1  = (const float*)d_in[5];
  const float* Wr1  = (const float*)d_in[6];   // [128,64]
  const float* g1   = (const float*)d_in[7];
  const float* be1  = (const float*)d_in[8];
  const float* Wl2  = (const float*)d_in[9];   // [64,32]
  const float* bl2  = (const float*)d_in[10];
  const float* Wr2  = (const float*)d_in[11];  // [64,32]
  const float* g2   = (const float*)d_in[12];
  const float* be2  = (const float*)d_in[13];
  const float* Wo1  = (const float*)d_in[14];  // [32,16]
  const float* bo1  = (const float*)d_in[15];
  const float* Wo2  = (const float*)d_in[16];  // [16,1]
  const float* bo2  = (const float*)d_in[17];

  const int* src = ei;            // edge_index[0]
  const int* tgt = ei + NEDGES;   // edge_index[1]

  // Workspace layout (floats)
  float* ws  = (float*)d_ws;
  float* h0  = ws;                         // 50000*128 = 6,400,000
  float* agg = ws + 6400000;               // 6,400,000 (cnt must follow)
  float* cnt = ws + 12800000;              // 50,000
  float* tmp = ws + 12850048;              // pre-LN, up to 50000*64 = 3,200,000
  float* h1  = ws + 16050048;              // 50000*64 = 3,200,000
  float* h2  = h0;                         // reuse h0 region: 50000*32
  float* h3  = h0 + 1600000;               // 50000*16 (disjoint from h2)

  const dim3 blk(256);

  // Stage 0: h0 = relu(x @ W_in + b_in)  [50000,128]; tiles = 3125*8
  gemm_bias_wmma<true><<<dim3((3125 * 8) / 8), blk, 0, stream>>>(
      x, W_in, b_in, h0, NNODES, 84, 128);

  // Stage 1: mean aggregation of h0 (D=128)
  zero_f32<<<dim3((6450000 + 255) / 256), blk, 0, stream>>>(agg, 6450000LL);
  scatter_add<<<dim3((int)(((long long)NEDGES * 128 + 255) / 256)), blk, 0, stream>>>(
      src, tgt, h0, agg, cnt, 7);
  // combine: tmp = agg/cnt @ Wl1 + bl1 + h0 @ Wr1   [50000,64]; tiles = 12500
  sage_combine_wmma<<<dim3((12500 + 7) / 8), blk, 0, stream>>>(
      agg, cnt, h0, Wl1, bl1, Wr1, tmp, NNODES, 128, 64);
  ln_relu<<<dim3((NNODES + 7) / 8), blk, 0, stream>>>(tmp, g1, be1, h1, 64);

  // Stage 2: mean aggregation of h1 (D=64)
  zero_f32<<<dim3((6450000 + 255) / 256), blk, 0, stream>>>(agg, 6450000LL);
  scatter_add<<<dim3((int)(((long long)NEDGES * 64 + 255) / 256)), blk, 0, stream>>>(
      src, tgt, h1, agg, cnt, 6);
  // combine: tmp = agg/cnt @ Wl2 + bl2 + h1 @ Wr2   [50000,32]; tiles = 6250
  sage_combine_wmma<<<dim3((6250 + 7) / 8), blk, 0, stream>>>(
      agg, cnt, h1, Wl2, bl2, Wr2, tmp, NNODES, 64, 32);
  ln_relu<<<dim3((NNODES + 7) / 8), blk, 0, stream>>>(tmp, g2, be2, h2, 32);

  // Output MLP: h3 = relu(h2 @ Wo1 + bo1) [50000,16]; tiles = 3125
  gemm_bias_wmma<true><<<dim3((3125 + 7) / 8), blk, 0, stream>>>(
      h2, Wo1, bo1, h3, NNODES, 32, 16);
  final_proj<<<dim3((NNODES + 255) / 256), blk, 0, stream>>>(
      h3, Wo2, bo2, (float*)d_out);
}